// DiscAdvLossForSource_PartialDA_1606317768947
// MI455X (gfx1250) — compile-verified
//
#include <hip/hip_runtime.h>
#include <hip/hip_bf16.h>
#include <math.h>

typedef float v2f __attribute__((ext_vector_type(2)));
typedef float v8f __attribute__((ext_vector_type(8)));

#define NCLS   2048
#define CHUNKS 16          // 2048 cols / (32 lanes * 4 floats)
#define EPSF   1e-6f
#define L2E    1.4426950408889634f   // log2(e)

// One wave32 per row. Streams the row once (global_load_b128), computes
// max + sum(exp(x-max)) in registers with a fused fma+v_exp_f32 per element,
// reduces the 32-lane exp-sum with V_WMMA_F32_16X16X4_F32 (B = ones =>
// every D column holds the total of A), then lane 0 computes the
// focal-loss term for this row.
__global__ __launch_bounds__(256) void focal_row_kernel(
    const float* __restrict__ in, const int* __restrict__ target,
    const float* __restrict__ class_weight, float* __restrict__ row_out)
{
    const int lane = threadIdx.x & 31;
    const int row  = blockIdx.x * 8 + (threadIdx.x >> 5);

    const float* rp = in + (size_t)row * NCLS;
    const float4* rp4 = reinterpret_cast<const float4*>(rp);

    float4 v[CHUNKS];
#pragma unroll
    for (int j = 0; j < CHUNKS; ++j)
        v[j] = rp4[j * 32 + lane];          // coalesced b128 stream

    // ---- row max ----
    float m = -INFINITY;
#pragma unroll
    for (int j = 0; j < CHUNKS; ++j) {
        m = fmaxf(m, fmaxf(fmaxf(v[j].x, v[j].y), fmaxf(v[j].z, v[j].w)));
    }
#pragma unroll
    for (int off = 16; off >= 1; off >>= 1)
        m = fmaxf(m, __shfl_xor(m, off, 32));

    // ---- per-lane partial sum of exp(x - m) ----
    // exp(x-m) = exp2(x*log2e - m*log2e): one v_fma + one v_exp per element.
    const float nml2e = -m * L2E;
    float s = 0.f;
#pragma unroll
    for (int j = 0; j < CHUNKS; ++j) {
        s += __builtin_amdgcn_exp2f(__builtin_fmaf(v[j].x, L2E, nml2e));
        s += __builtin_amdgcn_exp2f(__builtin_fmaf(v[j].y, L2E, nml2e));
        s += __builtin_amdgcn_exp2f(__builtin_fmaf(v[j].z, L2E, nml2e));
        s += __builtin_amdgcn_exp2f(__builtin_fmaf(v[j].w, L2E, nml2e));
    }

    // ---- 32-lane sum via WMMA f32 16x16x4 with all-ones B ----
    // With B all-ones, each D column's 16 entries partition the sum of all A
    // elements; summing my half's 8 D VGPRs and adding the opposite half
    // (xor 16) yields the full 32-lane sum in every lane.
    v2f a; a.x = s;   a.y = 0.f;
    v2f b; b.x = 1.f; b.y = 1.f;
    v8f c = {};
    v8f d = __builtin_amdgcn_wmma_f32_16x16x4_f32(
        /*neg_a=*/false, a, /*neg_b=*/false, b,
        /*c_mod=*/(short)0, c, /*reuse_a=*/false, /*reuse_b=*/false);
    float u = d[0] + d[1] + d[2] + d[3] + d[4] + d[5] + d[6] + d[7];
    float S = u + __shfl_xor(u, 16, 32);

    if (lane == 0) {
        const int   t  = target[row];
        const float xt = rp[t];          // L0/L2 hit: row just streamed
        const float xd = rp[NCLS - 1];
        const float w  = class_weight[t];

        const float pt = __builtin_amdgcn_exp2f(__builtin_fmaf(xt, L2E, nml2e)) / S;
        const float pd = __builtin_amdgcn_exp2f(__builtin_fmaf(xd, L2E, nml2e)) / S;

        // log(where(pt==0, pt+EPS, pt))
        const float log_pt   = __logf(pt == 0.f ? EPSF : pt);
        // log(1 - where(pt==1, (1-EPS)*pt, pt))  -> pt==1 gives log(EPS)
        const float log_1mpt = __logf(pt == 1.f ? EPSF : (1.f - pt));

        row_out[row] = w * (-log_pt * (1.f - pd) - log_1mpt * pd);
    }
}

// Deterministic final reduction: fixed strided order -> LDS tree -> shfl.
__global__ __launch_bounds__(256) void reduce_kernel(
    const float* __restrict__ row_vals, float* __restrict__ out, int n)
{
    __shared__ float sm[256];
    const int tid = threadIdx.x;

    float acc = 0.f;
    for (int i = tid; i < n; i += 256)
        acc += row_vals[i];
    sm[tid] = acc;
    __syncthreads();

    for (int stride = 128; stride >= 32; stride >>= 1) {
        if (tid < stride) sm[tid] += sm[tid + stride];
        __syncthreads();
    }
    if (tid < 32) {
        float vv = sm[tid];
#pragma unroll
        for (int off = 16; off >= 1; off >>= 1)
            vv += __shfl_xor(vv, off, 32);
        if (tid == 0)
            out[0] = vv / (float)n;
    }
}

extern "C" void kernel_launch(void* const* d_in, const int* in_sizes, int n_in,
                              void* d_out, int out_size, void* d_ws, size_t ws_size,
                              hipStream_t stream)
{
    const float* input        = (const float*)d_in[0];
    const int*   target       = (const int*)  d_in[1];
    const float* class_weight = (const float*)d_in[2];
    float*       out          = (float*)d_out;
    float*       row_vals     = (float*)d_ws;     // B floats of scratch

    const int B = in_sizes[1];                    // 32768 rows
    const int rows_per_block = 8;                 // 8 waves * 1 row/wave
    const int grid = B / rows_per_block;          // 4096

    focal_row_kernel<<<grid, 256, 0, stream>>>(input, target, class_weight, row_vals);
    reduce_kernel<<<1, 256, 0, stream>>>(row_vals, out, B);
}